// KLDivergence_21766894256272
// MI455X (gfx1250) — compile-verified
//
#include <hip/hip_runtime.h>

// KL divergence of empirical Gaussian vs N(0, I), reduced to per-column moments.
// Only the diagonal of Y^T Y is observable in the output, so we never form the
// Gram matrix: one streaming pass computes column sums / sums-of-squares using
// V_WMMA_F32_16X16X4_F32 with an all-ones A-matrix as the row-reduction engine.
//
// Stage 1 waves cover 32 columns via b64 (float2) loads so each 16-lane half
// reads a full 128B cacheline per row (pure-bandwidth kernel: ~5.8us floor at
// 23.3 TB/s for the 134MB input). Even/odd columns feed two independent WMMA
// accumulator sets; K-slot ordering inside the B layout is irrelevant for sums.

typedef __attribute__((ext_vector_type(2))) float v2f;
typedef __attribute__((ext_vector_type(8))) float v8f;

#define B_ROWS 8192
#define N_COLS 4096
#define ROWCHUNKS 64
#define ROWS_PER_CHUNK (B_ROWS / ROWCHUNKS)   // 128
#define WMMA_ITERS (ROWS_PER_CHUNK / 4)       // 32 (K=4 per WMMA)
#define COLTILE 32                            // columns per wave (2x16 interleaved)

// ---------------- Stage 1: per-(32-col tile, 128-row chunk) moments ----------
__global__ __launch_bounds__(32)
void colmoments_wmma_kernel(const float* __restrict__ X,
                            float* __restrict__ s1,
                            float* __restrict__ s2) {
    const int ct   = blockIdx.x;          // 32-column tile
    const int rc   = blockIdx.y;          // 128-row chunk
    const int lane = threadIdx.x;         // wave32
    const int half = lane >> 4;           // which k-pair this lane supplies
    const int n    = lane & 15;           // B-matrix column slot
    const int c0   = ct * COLTILE;        // first column of tile

    // Lane n handles columns c0+2n (even set) and c0+2n+1 (odd set).
    const float* colp = X + c0 + 2 * n;
    int row = rc * ROWS_PER_CHUNK + 2 * half;

    v2f a; a.x = 1.0f; a.y = 1.0f;        // all-ones A (16x4): D[m,n] = sum_k B[k,n]
    v8f acc1e = {}, acc2e = {};           // sum(x), sum(x^2): even columns
    v8f acc1o = {}, acc2o = {};           // sum(x), sum(x^2): odd columns

    #pragma unroll 4
    for (int it = 0; it < WMMA_ITERS; ++it) {
        // Two rows (k and k+1 for this lane), 2 adjacent columns each: b64 loads.
        v2f lo = *(const v2f*)(colp + (size_t)(row    ) * N_COLS);
        v2f hi = *(const v2f*)(colp + (size_t)(row + 1) * N_COLS);
        row += 4;

        v2f be; be.x = lo.x; be.y = hi.x;     // even column, two k-rows
        v2f bo; bo.x = lo.y; bo.y = hi.y;     // odd  column, two k-rows
        v2f be2 = be * be;
        v2f bo2 = bo * bo;

        // 8 args: (neg_a, A, neg_b, B, c_mod, C, reuse_a, reuse_b)
        acc1e = __builtin_amdgcn_wmma_f32_16x16x4_f32(
                    false, a, false, be,  (short)0, acc1e, false, false);
        acc2e = __builtin_amdgcn_wmma_f32_16x16x4_f32(
                    false, a, false, be2, (short)0, acc2e, false, false);
        acc1o = __builtin_amdgcn_wmma_f32_16x16x4_f32(
                    false, a, false, bo,  (short)0, acc1o, false, false);
        acc2o = __builtin_amdgcn_wmma_f32_16x16x4_f32(
                    false, a, false, bo2, (short)0, acc2o, false, false);
    }

    // C/D layout: VGPR0, lanes 0-15 hold (M=0, N=lane); all D rows identical.
    // Lane n repacks {even,odd} column sums into a coalesced b64 store.
    if (lane < 16) {
        v2f o1; o1.x = acc1e[0]; o1.y = acc1o[0];
        v2f o2; o2.x = acc2e[0]; o2.y = acc2o[0];
        *(v2f*)(s1 + (size_t)rc * N_COLS + c0 + 2 * n) = o1;
        *(v2f*)(s2 + (size_t)rc * N_COLS + c0 + 2 * n) = o2;
    }
}

// ------------- Stage 2: fold row-chunk partials -> per-column contribution ---
__global__ __launch_bounds__(128)
void column_contrib_kernel(const float* __restrict__ s1,
                           const float* __restrict__ s2,
                           float* __restrict__ contrib) {
    const int j = blockIdx.x * 128 + threadIdx.x;   // one column per thread
    float S1 = 0.0f, S2 = 0.0f;
    for (int rc = 0; rc < ROWCHUNKS; ++rc) {
        S1 += s1[(size_t)rc * N_COLS + j];
        S2 += s2[(size_t)rc * N_COLS + j];
    }
    float mean = S1 * (1.0f / (float)B_ROWS);
    float sc   = S2 - (float)B_ROWS * mean * mean;   // sum((x-mean)^2)
    sc         = fmaxf(sc, 1e-18f);                  // clip BEFORE /N (matches ref)
    float d    = sc * (1.0f / (float)N_COLS);
    contrib[j] = d - logf(d) + mean * mean;
}

// ------------- Stage 3: reduce 4096 contributions to the scalar --------------
__global__ __launch_bounds__(256)
void kl_finalize_kernel(const float* __restrict__ contrib,
                        float* __restrict__ out) {
    __shared__ float red[256];
    const int t = threadIdx.x;
    float local = 0.0f;
    for (int j = t; j < N_COLS; j += 256) local += contrib[j];
    red[t] = local;
    __syncthreads();
    for (int s = 128; s > 0; s >>= 1) {
        if (t < s) red[t] += red[t + s];
        __syncthreads();
    }
    if (t == 0) out[0] = 0.5f * (red[0] - (float)N_COLS);
}

extern "C" void kernel_launch(void* const* d_in, const int* in_sizes, int n_in,
                              void* d_out, int out_size, void* d_ws, size_t ws_size,
                              hipStream_t stream) {
    const float* X = (const float*)d_in[0];
    float* out = (float*)d_out;

    // Workspace: S1 partials [64][4096], S2 partials [64][4096], contrib [4096].
    // Every word read is written earlier in the same launch -> deterministic.
    float* s1      = (float*)d_ws;
    float* s2      = s1 + (size_t)ROWCHUNKS * N_COLS;
    float* contrib = s2 + (size_t)ROWCHUNKS * N_COLS;

    dim3 grid1(N_COLS / COLTILE, ROWCHUNKS);          // 128 x 64 single-wave blocks
    colmoments_wmma_kernel<<<grid1, 32, 0, stream>>>(X, s1, s2);
    column_contrib_kernel<<<N_COLS / 128, 128, 0, stream>>>(s1, s2, contrib);
    kl_finalize_kernel<<<1, 256, 0, stream>>>(contrib, out);
}